// WPMD_Block_87729001988371
// MI455X (gfx1250) — compile-verified
//
#include <hip/hip_runtime.h>
#include <hip/hip_bf16.h>

typedef __attribute__((ext_vector_type(2))) float v2f;
typedef __attribute__((ext_vector_type(8))) float v8f;

#define B_  2
#define C_  64
#define H_  512
#define W_  512
#define NO_ 259          // (512 + 8 - 1) / 2 : db4 subband length
#define TP  64           // pixels per block in fused kernel
#define PAD 65           // LDS row stride (floats) to avoid bank conflicts

// db4 analysis filters. DEC_LO = REC_LO reversed; DEC_HI[n] = (-1)^(n+1) REC_LO[n].
__constant__ float DEC_LO[8] = {
    -0.010597401784997278f,  0.032883011666982945f,  0.030841381835986965f,
    -0.18703481171888114f,  -0.02798376941698385f,   0.6308807679295904f,
     0.7148465705525415f,    0.23037781330885523f };
__constant__ float DEC_HI[8] = {
    -0.23037781330885523f,   0.7148465705525415f,   -0.6308807679295904f,
    -0.02798376941698385f,   0.18703481171888114f,   0.030841381835986965f,
    -0.032883011666982945f, -0.010597401784997278f };

__device__ __forceinline__ int symref(int t, int n) {
    if (t < 0)  t = -1 - t;
    if (t >= n) t = 2 * n - 1 - t;
    return t;
}

// ---------------- Kernel 1: db4 DWT along W -------------------------------
// One block per (b,c,h) row. Row staged in LDS; aW/dW are [B*C*H, NO_].
__global__ void __launch_bounds__(128)
dwt_w_kernel(const float* __restrict__ x, float* __restrict__ aW, float* __restrict__ dW) {
    __shared__ float row[W_];
    const int r = blockIdx.x;                    // 0 .. B*C*H-1
    const float* src = x + (size_t)r * W_;
    for (int i = threadIdx.x; i < W_; i += 128) row[i] = src[i];
    __syncthreads();
    float* ao = aW + (size_t)r * NO_;
    float* dq = dW + (size_t)r * NO_;
    for (int k = threadIdx.x; k < NO_; k += 128) {
        float a = 0.f, d = 0.f;
        #pragma unroll
        for (int m = 0; m < 8; ++m) {
            float v = row[symref(2 * k + 1 - m, W_)];
            a += DEC_LO[m] * v;
            d += DEC_HI[m] * v;
        }
        ao[k] = a;
        dq[k] = d;
    }
}

// ---------------- Kernel 2: db4 DWT along H + Perona-Malik ----------------
// One block per (b,c,h'); threads stride over w' (coalesced).
// u_LH = Hi-pass_H(aW), u_HL = Lo-pass_H(dW); outputs g*u_LH, g*u_HL [B*C, NO_, NO_].
__global__ void __launch_bounds__(256)
dwt_h_g_kernel(const float* __restrict__ aW, const float* __restrict__ dW,
               float* __restrict__ gLH, float* __restrict__ gHL) {
    const int bx = blockIdx.x;                   // (B*C)*NO_
    const int bc = bx / NO_;
    const int hp = bx - bc * NO_;
    const float* aB = aW + (size_t)bc * H_ * NO_;
    const float* dB = dW + (size_t)bc * H_ * NO_;
    float* lhO = gLH + ((size_t)bc * NO_ + hp) * NO_;
    float* hlO = gHL + ((size_t)bc * NO_ + hp) * NO_;
    int rows[8];
    #pragma unroll
    for (int m = 0; m < 8; ++m) rows[m] = symref(2 * hp + 1 - m, H_);
    for (int w = threadIdx.x; w < NO_; w += 256) {
        float lh = 0.f, hl = 0.f;
        #pragma unroll
        for (int m = 0; m < 8; ++m) {
            lh += DEC_HI[m] * aB[(size_t)rows[m] * NO_ + w];
            hl += DEC_LO[m] * dB[(size_t)rows[m] * NO_ + w];
        }
        float gm2 = lh * lh + hl * hl + 1e-8f;
        float g = 1.0f / (1.0f + gm2 * 100.0f);   // /= k^2 = 0.01
        lhO[w] = g * lh;
        hlO[w] = g * hl;
    }
}

// ---------------- Kernel 3: upsample+add -> WMMA GEMM -> LN -> ReLU -------
// Block: 128 threads (4 waves) handles one 64-pixel row segment (b, h, w0..w0+63).
// u_k tile [C=64][64px] in LDS; each wave does 16 pixels x 64 out-channels via
// v_wmma_f32_16x16x4_f32 (16 K-steps x 4 M-tiles).
__global__ void __launch_bounds__(128)
fused_up_gemm_ln_kernel(const float* __restrict__ x,
                        const float* __restrict__ gLH, const float* __restrict__ gHL,
                        const float* __restrict__ conv_w, const float* __restrict__ conv_b,
                        const float* __restrict__ ln_w, const float* __restrict__ ln_b,
                        float* __restrict__ out) {
    __shared__ float uk[C_][PAD];    // u_k tile, [in-channel][pixel]
    __shared__ float wl[C_][PAD];    // conv_w,   [out-channel][in-channel]
    __shared__ float yt[TP][PAD];    // y tile,   [pixel][out-channel]
    __shared__ float mu_s[TP];
    __shared__ float rs_s[TP];

    const int t  = threadIdx.x;
    const int bx = blockIdx.x;           // B * H * (W/TP) = 2*512*8
    const int wt = bx & 7;
    const int h  = (bx >> 3) & (H_ - 1);
    const int b  = bx >> 12;
    const int w0 = wt * TP;

    // stage weights
    for (int idx = t; idx < C_ * C_; idx += 128)
        wl[idx >> 6][idx & 63] = conv_w[idx];

    // bilinear source coords (half-pixel centers, edge clamp)
    const float S = (float)NO_ / (float)W_;
    float sh = ((float)h + 0.5f) * S - 0.5f;
    int h0 = (int)floorf(sh);
    float fh = sh - (float)h0;
    int h1 = min(max(h0 + 1, 0), NO_ - 1);
    h0 = min(max(h0, 0), NO_ - 1);

    // build u_k tile: x + up(g*u_LH) + up(g*u_HL)
    for (int idx = t; idx < C_ * TP; idx += 128) {
        int c = idx >> 6;
        int p = idx & 63;
        int w = w0 + p;
        float sw = ((float)w + 0.5f) * S - 0.5f;
        int q0 = (int)floorf(sw);
        float fw = sw - (float)q0;
        int q1 = min(max(q0 + 1, 0), NO_ - 1);
        q0 = min(max(q0, 0), NO_ - 1);
        const size_t pb = (size_t)(b * C_ + c) * NO_ * NO_;
        const float* L = gLH + pb;
        const float* Hp = gHL + pb;
        float w00 = (1.f - fh) * (1.f - fw), w01 = (1.f - fh) * fw;
        float w10 = fh * (1.f - fw),         w11 = fh * fw;
        size_t r0 = (size_t)h0 * NO_, r1 = (size_t)h1 * NO_;
        float lh = w00 * L[r0 + q0] + w01 * L[r0 + q1] + w10 * L[r1 + q0] + w11 * L[r1 + q1];
        float hl = w00 * Hp[r0 + q0] + w01 * Hp[r0 + q1] + w10 * Hp[r1 + q0] + w11 * Hp[r1 + q1];
        float xv = x[((size_t)(b * C_ + c) * H_ + h) * W_ + w];
        uk[c][p] = xv + lh + hl;
    }
    __syncthreads();

    // ---- WMMA GEMM: y[o][p] = sum_c W[o][c] * uk[c][p] ----
    // A (16x4 f32): lanes 0-15 -> M=lane, K = 4k+{0,1}; lanes 16-31 -> K = 4k+{2,3}.
    // B (4x16 f32): lanes 0-15 -> row K=4k+{0,1} at N=lane; lanes 16-31 -> K=4k+{2,3}.
    // C/D (16x16 f32): VGPR r -> (M=r, lanes 0-15) and (M=r+8, lanes 16-31).
    const int lane = t & 31;
    const int ln16 = lane & 15;
    const int hi   = lane >> 4;
    const int pix0 = (t >> 5) * 16;      // wave's 16-pixel slice

    v8f acc0 = {}, acc1 = {}, acc2 = {}, acc3 = {};
    for (int k = 0; k < 16; ++k) {
        const int kc = 4 * k + 2 * hi;
        v2f bm; bm.x = uk[kc][pix0 + ln16];  bm.y = uk[kc + 1][pix0 + ln16];
        v2f a0; a0.x = wl[ln16][kc];         a0.y = wl[ln16][kc + 1];
        v2f a1; a1.x = wl[16 + ln16][kc];    a1.y = wl[16 + ln16][kc + 1];
        v2f a2; a2.x = wl[32 + ln16][kc];    a2.y = wl[32 + ln16][kc + 1];
        v2f a3; a3.x = wl[48 + ln16][kc];    a3.y = wl[48 + ln16][kc + 1];
        acc0 = __builtin_amdgcn_wmma_f32_16x16x4_f32(false, a0, false, bm, (short)0, acc0, false, false);
        acc1 = __builtin_amdgcn_wmma_f32_16x16x4_f32(false, a1, false, bm, (short)0, acc1, false, false);
        acc2 = __builtin_amdgcn_wmma_f32_16x16x4_f32(false, a2, false, bm, (short)0, acc2, false, false);
        acc3 = __builtin_amdgcn_wmma_f32_16x16x4_f32(false, a3, false, bm, (short)0, acc3, false, false);
    }

    // spill y tile (with bias) to LDS, transposed to [pixel][channel] for LN
    {
        const int p = pix0 + ln16;
        #pragma unroll
        for (int r = 0; r < 8; ++r) {
            int o = r + hi * 8;
            yt[p][o]      = acc0[r] + conv_b[o];
            yt[p][o + 16] = acc1[r] + conv_b[o + 16];
            yt[p][o + 32] = acc2[r] + conv_b[o + 32];
            yt[p][o + 48] = acc3[r] + conv_b[o + 48];
        }
    }
    __syncthreads();

    // per-pixel LayerNorm stats over C=64 (biased variance, eps 1e-5)
    if (t < TP) {
        float mu = 0.f;
        #pragma unroll 8
        for (int c = 0; c < C_; ++c) mu += yt[t][c];
        mu *= (1.0f / C_);
        float var = 0.f;
        #pragma unroll 8
        for (int c = 0; c < C_; ++c) { float d = yt[t][c] - mu; var += d * d; }
        var *= (1.0f / C_);
        mu_s[t] = mu;
        rs_s[t] = rsqrtf(var + 1e-5f);
    }
    __syncthreads();

    // normalize + affine + ReLU, coalesced store (consecutive lanes -> consecutive w)
    {
        const int p  = t & 63;
        const int c0 = t >> 6;
        const float mu = mu_s[p];
        const float rs = rs_s[p];
        for (int c = c0; c < C_; c += 2) {
            float v = (yt[p][c] - mu) * rs * ln_w[c] + ln_b[c];
            v = fmaxf(v, 0.0f);
            out[((size_t)(b * C_ + c) * H_ + h) * W_ + w0 + p] = v;
        }
    }
}

extern "C" void kernel_launch(void* const* d_in, const int* in_sizes, int n_in,
                              void* d_out, int out_size, void* d_ws, size_t ws_size,
                              hipStream_t stream) {
    const float* x      = (const float*)d_in[0];
    const float* conv_w = (const float*)d_in[1];
    const float* conv_b = (const float*)d_in[2];
    const float* ln_w   = (const float*)d_in[3];
    const float* ln_b   = (const float*)d_in[4];
    float* out = (float*)d_out;

    // workspace layout (floats): aW | dW : [B*C*H, NO_]; gLH | gHL : [B*C, NO_, NO_]
    const size_t rowElems = (size_t)B_ * C_ * H_ * NO_;   // 16,973,824
    const size_t subElems = (size_t)B_ * C_ * NO_ * NO_;  //  8,586,368
    float* aW  = (float*)d_ws;
    float* dW  = aW + rowElems;
    float* gLH = dW + rowElems;
    float* gHL = gLH + subElems;

    dwt_w_kernel<<<B_ * C_ * H_, 128, 0, stream>>>(x, aW, dW);
    dwt_h_g_kernel<<<B_ * C_ * NO_, 256, 0, stream>>>(aW, dW, gLH, gHL);
    fused_up_gemm_ln_kernel<<<B_ * H_ * (W_ / TP), 128, 0, stream>>>(
        x, gLH, gHL, conv_w, conv_b, ln_w, ln_b, out);
}